// PathSimplifiedGCN_2001454760614
// MI455X (gfx1250) — compile-verified
//
#include <hip/hip_runtime.h>

// PathSimplifiedGCN on MI455X (gfx1250, wave32).
// Exact linearity rewrite: pooled_0=(A@x)@W0+s*b0; pooled_i=pooled_{i-1}@Wi+s*bi.
// Heavy step P=A@x is an L2-resident gather/accumulate; the chained 128x512x512
// GEMMs use V_WMMA_F32_16X16X4_F32 (f32 WMMA path, matching reference dtype).

#define N_NODES   50000
#define N_GRAPHS  128
#define HID       512
#define OUT_PITCH (3 * HID)
#define MAX_CHUNKS 64
#define COLS_PER_WG 256   // column slice per workgroup in accumulate kernel

typedef __attribute__((ext_vector_type(2))) float v2f;
typedef __attribute__((ext_vector_type(8))) float v8f;

// ---------------------------------------------------------------- zero counts
__global__ void k_zero_counts(unsigned* __restrict__ counts) {
    counts[threadIdx.x] = 0u;
}

// ------------------------------------------- path_batch gather + histogram
__global__ void k_pb_hist(const int* __restrict__ walk2_row0,
                          const int* __restrict__ batch,
                          int* __restrict__ pb,
                          unsigned* __restrict__ counts) {
    int n = blockIdx.x * 256 + threadIdx.x;
    if (n < N_NODES) {
        int g = batch[walk2_row0[n]];   // path_batch[n] = batch[walk2[0][n]]
        pb[n] = g;
        atomicAdd(&counts[g], 1u);      // integer atomics: order-independent
    }
}

// ------------------------------------------------- partial A@x accumulation
// grid = (nchunks, 2). Each WG owns positions [n0,n1) and a 256-column slice.
// Thread t owns exactly one column -> deterministic sequential accumulation.
__global__ void __launch_bounds__(256)
k_accum(const float* __restrict__ x,
        const int* __restrict__ w2, const int* __restrict__ w3,
        const int* __restrict__ w4, const int* __restrict__ pb,
        float* __restrict__ partial, int nchunks) {
    __shared__ float acc[N_GRAPHS * COLS_PER_WG];   // 128 KB LDS (<=320 KB/WGP)
    const int chunk = blockIdx.x;
    const int cg    = blockIdx.y;              // 0 or 1
    const int t     = threadIdx.x;             // 0..255
    const int c     = cg * COLS_PER_WG + t;    // global column

    for (int i = t; i < N_GRAPHS * COLS_PER_WG; i += 256) acc[i] = 0.0f;
    __syncthreads();

    const int per = (N_NODES + nchunks - 1) / nchunks;
    const int n0  = chunk * per;
    const int n1  = (n0 + per < N_NODES) ? (n0 + per) : N_NODES;

    for (int n = n0; n < n1; ++n) {
        const int g = pb[n];
        float y = 0.0f;
        #pragma unroll
        for (int l = 0; l < 3; ++l) y += x[w2[l * N_NODES + n] * HID + c];
        #pragma unroll
        for (int l = 0; l < 4; ++l) y += x[w3[l * N_NODES + n] * HID + c];
        #pragma unroll
        for (int l = 0; l < 5; ++l) y += x[w4[l * N_NODES + n] * HID + c];
        acc[g * COLS_PER_WG + t] += y;     // exclusive column, fixed order
    }
    __syncthreads();

    float* dst = partial + (size_t)chunk * (N_GRAPHS * HID) + cg * COLS_PER_WG;
    for (int g = 0; g < N_GRAPHS; ++g)
        dst[g * HID + t] = acc[g * COLS_PER_WG + t];
}

// ------------------------- deterministic partial reduce + denom + s vector
__global__ void k_reduce(const float* __restrict__ partial,
                         const unsigned* __restrict__ counts,
                         float* __restrict__ P, float* __restrict__ s,
                         int nchunks) {
    int i = blockIdx.x * 256 + threadIdx.x;    // 0 .. 128*512-1
    int g = i >> 9;                            // i / HID
    float a = 0.0f;
    for (int ch = 0; ch < nchunks; ++ch)       // fixed order
        a += partial[(size_t)ch * (N_GRAPHS * HID) + i];
    float cnt = (float)counts[g];
    P[i] = a / fmaxf(cnt, 1.0f);
    if (i < N_GRAPHS) s[i] = (counts[i] != 0u) ? 12.0f : 0.0f;
}

// ---------------------------------------- 128x512x512 GEMM via f32 WMMA
// out_tile = A(128xK=512, pitch lda) @ W(512x512) + s[m]*bias[n]
// One wave per 16x16 tile; 32 blocks * 8 waves = 256 tiles.
__global__ void __launch_bounds__(256)
k_gemm_wmma(const float* __restrict__ A, int lda,
            const float* __restrict__ W, const float* __restrict__ bias,
            const float* __restrict__ s, float* __restrict__ out) {
    const int wave = (blockIdx.x << 3) + (threadIdx.x >> 5);
    const int lane = threadIdx.x & 31;
    const int l16  = lane & 15;
    const int lh   = lane >> 4;          // half-wave select
    const int tm   = wave >> 5;          // 0..7   (M tile)
    const int tn   = wave & 31;          // 0..31  (N tile)

    const float* Arow = A + (tm * 16 + l16) * lda;   // A frag: lane = M row
    const int col = tn * 16 + l16;                   // B/D frag: lane = N col

    v8f acc = {};
    for (int k = 0; k < HID; k += 4) {
        const int kk = k + lh * 2;       // K pair per half-wave
        v2f a; a.x = Arow[kk];           a.y = Arow[kk + 1];
        v2f b; b.x = W[kk * HID + col];  b.y = W[(kk + 1) * HID + col];
        // D = A(16x4) * B(4x16) + C ; emits v_wmma_f32_16x16x4_f32
        acc = __builtin_amdgcn_wmma_f32_16x16x4_f32(
            /*neg_a=*/false, a, /*neg_b=*/false, b,
            /*c_mod=*/(short)0, acc, /*reuse_a=*/false, /*reuse_b=*/false);
    }

    const float bn = bias[col];
    #pragma unroll
    for (int v = 0; v < 8; ++v) {
        const int m = tm * 16 + v + lh * 8;          // C/D layout: M=v (+8 hi)
        out[m * OUT_PITCH + col] = acc[v] + s[m] * bn;
    }
}

// --------------------------------------------------------------------------
extern "C" void kernel_launch(void* const* d_in, const int* in_sizes, int n_in,
                              void* d_out, int out_size, void* d_ws, size_t ws_size,
                              hipStream_t stream) {
    const float* x     = (const float*)d_in[0];
    const int*   w2    = (const int*)  d_in[1];   // [3, N]
    const int*   w3    = (const int*)  d_in[2];   // [4, N]
    const int*   w4    = (const int*)  d_in[3];   // [5, N]
    const int*   batch = (const int*)  d_in[4];   // [N]
    const float* W0 = (const float*)d_in[5];  const float* b0 = (const float*)d_in[6];
    const float* W1 = (const float*)d_in[7];  const float* b1 = (const float*)d_in[8];
    const float* W2 = (const float*)d_in[9];  const float* b2 = (const float*)d_in[10];
    float* out = (float*)d_out;                   // [128, 1536] row-major

    // workspace carving (all 4-byte aligned)
    const size_t partBytes = (size_t)N_GRAPHS * HID * sizeof(float);   // 256 KB
    const size_t fixed = partBytes /*P*/ + 512 /*counts*/ + 512 /*s*/
                       + (size_t)N_NODES * sizeof(int) /*pb*/;
    int nchunks = 1;
    if (ws_size > fixed + partBytes) {
        size_t fit = (ws_size - fixed) / partBytes;
        nchunks = (fit > MAX_CHUNKS) ? MAX_CHUNKS : (int)fit;
        if (nchunks < 1) nchunks = 1;
    }
    char* ws = (char*)d_ws;
    float*    partial = (float*)ws;
    float*    P       = (float*)(ws + (size_t)nchunks * partBytes);
    unsigned* counts  = (unsigned*)((char*)P + partBytes);
    float*    s       = (float*)((char*)counts + 512);
    int*      pb      = (int*)((char*)s + 512);

    k_zero_counts<<<1, 128, 0, stream>>>(counts);
    k_pb_hist<<<(N_NODES + 255) / 256, 256, 0, stream>>>(w2, batch, pb, counts);

    dim3 gAcc(nchunks, 2);
    k_accum<<<gAcc, 256, 0, stream>>>(x, w2, w3, w4, pb, partial, nchunks);
    k_reduce<<<(N_GRAPHS * HID) / 256, 256, 0, stream>>>(partial, counts, P, s, nchunks);

    // chained pooled GEMMs (each depends on the previous -> stream order)
    k_gemm_wmma<<<32, 256, 0, stream>>>(P,         HID,       W0, b0, s, out);
    k_gemm_wmma<<<32, 256, 0, stream>>>(out,       OUT_PITCH, W1, b1, s, out + HID);
    k_gemm_wmma<<<32, 256, 0, stream>>>(out + HID, OUT_PITCH, W2, b2, s, out + 2 * HID);
}